// QNNUnsplitted_28338194219198
// MI455X (gfx1250) — compile-verified
//
#include <hip/hip_runtime.h>
#include <hip/hip_bf16.h>
#include <math.h>

typedef __attribute__((ext_vector_type(2))) float v2f;
typedef __attribute__((ext_vector_type(8))) float v8f;

#define NQ 8
#define DIM 256
#define DIN 1024
#define HID 512
#define DOUT 128
#define BATCH 32768
#define TSAMP 64
#define THETA1 1.0471975511965976f /* pi/3 */

// workspace float offsets
#define WS_WEFF 0              // 8*1024
#define WS_BEFF 8192           // 8 (+pad)
#define WS_MT   8256           // 256*512  (Mt[k][0..255]=Re U[., k], [256..511]=Im)
#define WS_WOUTT (8256+131072) // 256*128  (WoutT[k][o] = Wout[o][k])

// ---------------------------------------------------------------------------
// Kernel 1: fold linear layers, transpose Wout.
// ---------------------------------------------------------------------------
__global__ void prep_kernel(const float* __restrict__ W1, const float* __restrict__ b1,
                            const float* __restrict__ W2, const float* __restrict__ b2,
                            const float* __restrict__ scale_in,
                            const float* __restrict__ Wout,
                            float* __restrict__ ws) {
  int g = blockIdx.x * blockDim.x + threadIdx.x;
  float scale = scale_in[0];
  if (g < 8 * DIN) {
    int a = g >> 10, i = g & (DIN - 1);
    float acc = 0.f;
    for (int h = 0; h < HID; ++h)
      acc = fmaf(W2[a * HID + h], W1[h * DIN + i], acc);
    ws[WS_WEFF + g] = acc * scale;
  } else if (g < 8 * DIN + DIM * DOUT) {
    int gw = g - 8 * DIN;
    int k = gw >> 7, o = gw & (DOUT - 1);
    ws[WS_WOUTT + k * DOUT + o] = Wout[o * DIM + k];
  } else if (g < 8 * DIN + DIM * DOUT + 8) {
    int a = g - (8 * DIN + DIM * DOUT);
    float acc = 0.f;
    for (int h = 0; h < HID; ++h) acc = fmaf(W2[a * HID + h], b1[h], acc);
    ws[WS_BEFF + a] = (acc + b2[a]) * scale;
  }
}

// ---------------------------------------------------------------------------
// Kernel 2: build U (256x256 complex) by simulating the entangling circuit
// on each basis state. One workgroup per basis column k; state in LDS.
// Wire q <-> bit (7-q) (wire 0 = MSB, PennyLane order).
// ---------------------------------------------------------------------------
__global__ void build_u_kernel(const float* __restrict__ qweights, float* __restrict__ ws) {
  __shared__ float2 bufA[DIM];
  __shared__ float2 bufB[DIM];
  const int i = threadIdx.x;
  const int k = blockIdx.x;
  float2* cur = bufA;
  float2* nxt = bufB;
  cur[i] = (i == k) ? make_float2(1.f, 0.f) : make_float2(0.f, 0.f);
  __syncthreads();

  auto cmul = [](float2 a, float2 b) {
    return make_float2(a.x * b.x - a.y * b.y, a.x * b.y + a.y * b.x);
  };
  auto apply1q = [&](int q, float2 g00, float2 g01, float2 g10, float2 g11) {
    int b = 7 - q, m = 1 << b;
    float2 x0 = cur[i & ~m];
    float2 x1 = cur[i | m];
    int bit = (i >> b) & 1;
    float2 ga = bit ? g10 : g00;
    float2 gb = bit ? g11 : g01;
    float2 r0 = cmul(ga, x0);
    float2 r1 = cmul(gb, x1);
    nxt[i] = make_float2(r0.x + r1.x, r0.y + r1.y);
    __syncthreads();
    float2* t = cur; cur = nxt; nxt = t;
  };
  auto rx = [&](int q, float th) {
    float c = cosf(0.5f * th), s = sinf(0.5f * th);
    apply1q(q, make_float2(c, 0.f), make_float2(0.f, -s),
               make_float2(0.f, -s), make_float2(c, 0.f));
  };
  auto ry = [&](int q, float th) {
    float c = cosf(0.5f * th), s = sinf(0.5f * th);
    apply1q(q, make_float2(c, 0.f), make_float2(-s, 0.f),
               make_float2(s, 0.f), make_float2(c, 0.f));
  };
  auto cnot = [&](int c, int t) {
    int bc = 7 - c, bt = 7 - t;
    int src = ((i >> bc) & 1) ? (i ^ (1 << bt)) : i;
    nxt[i] = cur[src];
    __syncthreads();
    float2* tt = cur; cur = nxt; nxt = tt;
  };

  const float HPI = 1.5707963267948966f;
  for (int q = 0; q < NQ - 1; ++q) {
    int j = q + 1, p = 3 * q;
    rx(j, -HPI);
    cnot(j, q);
    rx(q, qweights[p] * THETA1);
    ry(j, qweights[p + 1] * THETA1);
    cnot(q, j);
    ry(j, qweights[p + 2] * THETA1);
    cnot(j, q);
    rx(q, HPI);
  }

  float* Mt = ws + WS_MT;
  Mt[k * 512 + i]       = cur[i].x;  // Re U[i][k]
  Mt[k * 512 + 256 + i] = cur[i].y;  // Im U[i][k]
}

// ---------------------------------------------------------------------------
// Kernel 3: fused per-tile pipeline. 256 threads = 8 wave32, 64 samples/WG.
// ---------------------------------------------------------------------------
__global__ void __launch_bounds__(256, 1) main_kernel(
    const float* __restrict__ inputs, const float* __restrict__ bout,
    const float* __restrict__ ws, float* __restrict__ out) {
  __shared__ float sWeff[8 * DIN];     // 32 KB
  __shared__ float sBeff[8];
  __shared__ float sCS[TSAMP * 16];    // 4 KB : [s][q][cos,sin]
  __shared__ float sPsi[TSAMP * DIM];  // 64 KB : psi tile, later reused for probs

  const int tid  = threadIdx.x;
  const int lane = tid & 31;
  const int wave = tid >> 5;
  const int wgBase = blockIdx.x * TSAMP;
  const float* Mt    = ws + WS_MT;
  const float* WoutT = ws + WS_WOUTT;

  for (int idx = tid; idx < 8 * DIN; idx += 256) sWeff[idx] = ws[WS_WEFF + idx];
  if (tid < 8) sBeff[tid] = ws[WS_BEFF + tid];
  __syncthreads();

  // ---- Step 1: angles -> half-angle cos/sin (8 samples per wave) ----
  for (int t = 0; t < 8; ++t) {
    const int sl = wave * 8 + t;
    const float4* xrow = (const float4*)(inputs + (size_t)(wgBase + sl) * DIN);
    float acc[8] = {0.f, 0.f, 0.f, 0.f, 0.f, 0.f, 0.f, 0.f};
#pragma unroll
    for (int it = 0; it < 8; ++it) {
      const float4 x = xrow[it * 32 + lane];
      const int ibase = (it * 32 + lane) * 4;
#pragma unroll
      for (int a = 0; a < 8; ++a) {
        const float4 wv = *(const float4*)&sWeff[a * DIN + ibase];
        acc[a] = fmaf(x.x, wv.x, acc[a]);
        acc[a] = fmaf(x.y, wv.y, acc[a]);
        acc[a] = fmaf(x.z, wv.z, acc[a]);
        acc[a] = fmaf(x.w, wv.w, acc[a]);
      }
    }
#pragma unroll
    for (int a = 0; a < 8; ++a) {
#pragma unroll
      for (int off = 16; off > 0; off >>= 1)
        acc[a] += __shfl_xor(acc[a], off, 32);
    }
    if (lane == 0) {
#pragma unroll
      for (int a = 0; a < 8; ++a) {
        float h = 0.5f * (acc[a] + sBeff[a]) + 0.78539816339744831f; // θ/2 + π/4
        sCS[sl * 16 + 2 * a]     = cosf(h);
        sCS[sl * 16 + 2 * a + 1] = sinf(h);
      }
    }
  }
  __syncthreads();

  // ---- Step 2: psi tile (rank-1 tensor product, real) ----
  for (int idx = tid; idx < TSAMP * DIM; idx += 256) {
    const int sl = idx >> 8, ii = idx & 255;
    float p = 1.f;
#pragma unroll
    for (int q = 0; q < 8; ++q) {
      const int bit = (ii >> (7 - q)) & 1;
      p *= sCS[sl * 16 + 2 * q + bit];
    }
    sPsi[idx] = p;
  }
  __syncthreads();

  // ---- Step 3+4: Y = Psi * M^T (f32 WMMA), probs = yr^2 + yi^2 ----
  const int m    = wave & 3;    // M-tile (16 samples)
  const int pgrp = wave >> 2;   // 0/1: which 8 paired N-tiles
  const int rowA = m * 16 + (lane & 15);
  const int koff = (lane >> 4) * 2;
  v8f fp[8];

#pragma unroll
  for (int pi = 0; pi < 8; ++pi) {
    const int pt   = pgrp * 8 + pi;
    const int colR = pt * 16 + (lane & 15);
    const int colI = colR + 256;
    v8f accR = {};
    v8f accI = {};
    for (int k0 = 0; k0 < DIM; k0 += 4) {
      v2f a;
      const float* ap = &sPsi[rowA * DIM + k0 + koff];
      a.x = ap[0]; a.y = ap[1];
      const float* bp = &Mt[(size_t)(k0 + koff) * 512];
      v2f bR, bI;
      bR.x = bp[colR]; bR.y = bp[512 + colR];
      bI.x = bp[colI]; bI.y = bp[512 + colI];
      accR = __builtin_amdgcn_wmma_f32_16x16x4_f32(false, a, false, bR, (short)0, accR, false, false);
      accI = __builtin_amdgcn_wmma_f32_16x16x4_f32(false, a, false, bI, (short)0, accI, false, false);
    }
    fp[pi] = accR * accR + accI * accI;
  }
  __syncthreads();  // everyone done reading Psi

  {
    const int rbase = m * 16 + (lane >> 4) * 8;
#pragma unroll
    for (int pi = 0; pi < 8; ++pi) {
      const int col = (pgrp * 8 + pi) * 16 + (lane & 15);
#pragma unroll
      for (int r = 0; r < 8; ++r)
        sPsi[(rbase + r) * DIM + col] = fp[pi][r];  // probs tile overwrites psi
    }
  }
  __syncthreads();

  // ---- Step 5: out = probs @ Wout^T + bout (f32 WMMA) ----
#pragma unroll
  for (int jj = 0; jj < 4; ++jj) {
    const int nn  = pgrp * 4 + jj;
    const int col = nn * 16 + (lane & 15);
    v8f acc = {};
    for (int k0 = 0; k0 < DIM; k0 += 4) {
      v2f a;
      const float* ap = &sPsi[rowA * DIM + k0 + koff];
      a.x = ap[0]; a.y = ap[1];
      const float* bp = &WoutT[(k0 + koff) * DOUT + col];
      v2f b;
      b.x = bp[0]; b.y = bp[DOUT];
      acc = __builtin_amdgcn_wmma_f32_16x16x4_f32(false, a, false, b, (short)0, acc, false, false);
    }
    const float bo = bout[col];
    const int rbase = wgBase + m * 16 + (lane >> 4) * 8;
#pragma unroll
    for (int r = 0; r < 8; ++r)
      out[(size_t)(rbase + r) * DOUT + col] = acc[r] + bo;
  }
}

// ---------------------------------------------------------------------------
extern "C" void kernel_launch(void* const* d_in, const int* in_sizes, int n_in,
                              void* d_out, int out_size, void* d_ws, size_t ws_size,
                              hipStream_t stream) {
  const float* inputs   = (const float*)d_in[0];
  const float* W1       = (const float*)d_in[1];
  const float* b1       = (const float*)d_in[2];
  const float* W2       = (const float*)d_in[3];
  const float* b2       = (const float*)d_in[4];
  const float* scale_in = (const float*)d_in[5];
  const float* qweights = (const float*)d_in[6];
  const float* Wout     = (const float*)d_in[7];
  const float* bout     = (const float*)d_in[8];
  float* out = (float*)d_out;
  float* ws  = (float*)d_ws;

  prep_kernel<<<161, 256, 0, stream>>>(W1, b1, W2, b2, scale_in, Wout, ws);
  build_u_kernel<<<DIM, 256, 0, stream>>>(qweights, ws);
  main_kernel<<<BATCH / TSAMP, 256, 0, stream>>>(inputs, bout, ws, out);
}